// HistogramLoss_31361851195672
// MI455X (gfx1250) — compile-verified
//
#include <hip/hip_runtime.h>
#include <hip/hip_bf16.h>

#define H     2048
#define HH    (H * H)            // 1 << 22
#define NIDX  2097152
#define BINS  256
#define TOTAL (3 * HH)           // 12,582,912 ; divisible by 64

typedef __attribute__((ext_vector_type(2))) float v2f;
typedef __attribute__((ext_vector_type(8))) float v8f;

// -------- workspace layout, in u32 words --------
// [0, 1536)            : histograms  hist_dst[3][256] , hist_ref[3][256]
// [1536, 2304)         : transfer tables  table[3][256]  (int32)
// [2304, 2304+131072)  : per-pixel "scattered" bitset (HH / 32 words)
// [WS_ACCUM]           : float loss accumulator
#define WS_HIST   0
#define WS_TABLE  1536
#define WS_FLAGS  2304
#define WS_ACCUM  (WS_FLAGS + (HH / 32))
#define WS_WORDS  (WS_ACCUM + 1)

// ---------------- kernel 0: zero workspace ----------------
__global__ void hm_zero_ws(unsigned* __restrict__ ws) {
    for (unsigned i = blockIdx.x * blockDim.x + threadIdx.x; i < WS_WORDS;
         i += gridDim.x * blockDim.x)
        ws[i] = 0u;
}

// ---------------- kernel 1: gathered histograms (LDS-privatized) ----------------
// Random gathers over ref/target/masks: whole working set (~132 MB) fits in the
// 192 MB L2, so this phase is L2-resident after first touch.
__global__ void hm_hist(const float* __restrict__ ref,
                        const float* __restrict__ tgt,
                        const float* __restrict__ msrc,
                        const float* __restrict__ mtar,
                        const int* __restrict__ i0, const int* __restrict__ i1,
                        const int* __restrict__ i2, const int* __restrict__ i3,
                        unsigned* __restrict__ ws) {
    __shared__ unsigned lh[6 * BINS];
    for (int t = threadIdx.x; t < 6 * BINS; t += blockDim.x) lh[t] = 0u;
    __syncthreads();

    for (int k = blockIdx.x * blockDim.x + threadIdx.x; k < NIDX;
         k += gridDim.x * blockDim.x) {
        const int pd = i0[k] * H + i1[k];     // dst sample position (ref image)
        const int pr = i2[k] * H + i3[k];     // ref sample position (target image)
        const float ms = msrc[pd];
        const float mt = mtar[pr];
#pragma unroll
        for (int c = 0; c < 3; ++c) {
            float rv = fmaf(ref[c * HH + pd], 0.5f, 0.5f) * 255.0f * ms;
            int bd = min(max((int)floorf(rv), 0), BINS - 1);
            atomicAdd(&lh[c * BINS + bd], 1u);
            float tv = fmaf(tgt[c * HH + pr], 0.5f, 0.5f) * 255.0f * mt;
            int br = min(max((int)floorf(tv), 0), BINS - 1);
            atomicAdd(&lh[(3 + c) * BINS + br], 1u);
        }
    }
    __syncthreads();

    unsigned* hist = ws + WS_HIST;
    for (int t = threadIdx.x; t < 6 * BINS; t += blockDim.x) {
        unsigned v = lh[t];
        if (v) atomicAdd(&hist[t], v);
    }
}

// ---------------- kernel 2: CDFs + transfer tables (1 block, 256 threads) ----------------
__global__ void hm_table(unsigned* __restrict__ ws) {
    __shared__ float cd[BINS];
    __shared__ float cr[BINS];
    const unsigned* hist = ws + WS_HIST;
    int* table = (int*)(ws + WS_TABLE);
    const int t = threadIdx.x;

    for (int c = 0; c < 3; ++c) {
        cd[t] = (float)hist[c * BINS + t];
        cr[t] = (float)hist[(3 + c) * BINS + t];
        __syncthreads();
        if (t == 0) {  // 256-element sequential cumsum — negligible
            float a = 0.f, b = 0.f;
            for (int i = 0; i < BINS; ++i) {
                a += cd[i]; cd[i] = a;
                b += cr[i]; cr[i] = b;
            }
        }
        __syncthreads();
        const float r    = cd[t] / cd[BINS - 1];     // cdf_dst[t]
        const float mycr = cr[t] / cr[BINS - 1];     // cdf_ref[t]
        __syncthreads();
        cr[t] = mycr;
        __syncthreads();

        int tbl = t;                                  // "not found" -> identity
        for (int j = 1; j < BINS; ++j) {
            if (cr[j - 1] <= r && r <= cr[j]) { tbl = j; break; }
        }
        if (t == 0)        tbl = 0;
        if (t == BINS - 1) tbl = BINS - 1;
        table[c * BINS + t] = tbl;
        __syncthreads();
    }
}

// ---------------- kernel 3: scatter "was-matched" bitset ----------------
__global__ void hm_flags(const int* __restrict__ i0, const int* __restrict__ i1,
                         unsigned* __restrict__ ws) {
    unsigned* flags = ws + WS_FLAGS;
    for (int k = blockIdx.x * blockDim.x + threadIdx.x; k < NIDX;
         k += gridDim.x * blockDim.x) {
        unsigned p = (unsigned)(i0[k] * H + i1[k]);
        atomicOr(&flags[p >> 5], 1u << (p & 31u));
    }
}

// ---------------- kernel 4: streaming |diff| reduction via V_WMMA_F32_16X16X4_F32 ----------------
// Each wave feeds 64 f32 |diff| values per step into B (4x16); A = ones(16x4),
// so D = A*B + C accumulates exact f32 column sums; grand total = sum(C)/16.
// Wave id is readfirstlane'd -> scalar loop, EXEC stays all-ones (WMMA requirement).
// Per lane-iteration: b64 loads of inp/ref/msrc pairs, one flags word, branchless
// table select. e0 is even and plane size 2^22 is even, so both elements of a
// lane share the channel plane and the flags word.
__global__ void hm_loss(const float* __restrict__ inp,
                        const float* __restrict__ ref,
                        const float* __restrict__ msrc,
                        unsigned* __restrict__ ws) {
    const unsigned* flags = ws + WS_FLAGS;
    const int* table = (const int*)(ws + WS_TABLE);
    float* accum = (float*)(ws + WS_ACCUM);

    const unsigned lane   = threadIdx.x & 31u;
    const unsigned wave   = (unsigned)__builtin_amdgcn_readfirstlane(
        (int)(blockIdx.x * (blockDim.x >> 5) + (threadIdx.x >> 5)));
    const unsigned nwaves = gridDim.x * (blockDim.x >> 5);
    const unsigned NCHUNK = TOTAL / 64u;               // 196608, no tail
    const unsigned estride = nwaves * 64u;             // element stride per wave step

    v8f c = {0.f, 0.f, 0.f, 0.f, 0.f, 0.f, 0.f, 0.f};
    const v2f a = {1.0f, 1.0f};                        // ones A-matrix (16x4)

    for (unsigned chunk = wave; chunk < NCHUNK; chunk += nwaves) {
        const unsigned e0 = chunk * 64u + lane * 2u;
        const unsigned ch = e0 >> 22;                  // HH == 1<<22; uniform in wave
        const unsigned p0 = e0 & (HH - 1u);

        // prefetch one wave-stride ahead (emits global_prefetch_b8)
        if (e0 + estride < (unsigned)TOTAL) {
            __builtin_prefetch(inp + e0 + estride, 0, 0);
            __builtin_prefetch(ref + e0 + estride, 0, 0);
        }

        const float2 iv2 = *(const float2*)(inp + e0);
        const float2 rv2 = *(const float2*)(ref + e0);
        const float2 m2  = *(const float2*)(msrc + p0);
        const unsigned fw = flags[p0 >> 5];            // bits (p0&31) and (p0&31)+1

        v2f b;
        {   // element e0
            const float m  = m2.x;
            const float iv = fmaf(iv2.x, 0.5f, 0.5f) * 255.0f * m;
            const float rv = fmaf(rv2.x, 0.5f, 0.5f) * 255.0f * m;
            const int bin  = min(max((int)floorf(rv), 0), BINS - 1);
            const float tv = (float)table[ch * BINS + bin];
            const bool fl  = (fw >> (p0 & 31u)) & 1u;
            const float mv = (fl ? tv : rv) * m;
            b[0] = fabsf(iv - mv);
        }
        {   // element e0 + 1
            const float m  = m2.y;
            const float iv = fmaf(iv2.y, 0.5f, 0.5f) * 255.0f * m;
            const float rv = fmaf(rv2.y, 0.5f, 0.5f) * 255.0f * m;
            const int bin  = min(max((int)floorf(rv), 0), BINS - 1);
            const float tv = (float)table[ch * BINS + bin];
            const bool fl  = (fw >> ((p0 & 31u) + 1u)) & 1u;
            const float mv = (fl ? tv : rv) * m;
            b[1] = fabsf(iv - mv);
        }
        // EXEC all-ones: loop control is scalar (SGPR wave id), blocks are full waves.
        c = __builtin_amdgcn_wmma_f32_16x16x4_f32(false, a, false, b,
                                                  (short)0, c, false, false);
    }

    float s = c[0] + c[1] + c[2] + c[3] + c[4] + c[5] + c[6] + c[7];
#pragma unroll
    for (int off = 16; off > 0; off >>= 1) s += __shfl_xor(s, off, 32);
    if (lane == 0) atomicAdd(accum, s * (1.0f / 16.0f));   // rows replicate sums x16
}

// ---------------- kernel 5: finalize mean ----------------
__global__ void hm_final(const unsigned* __restrict__ ws, float* __restrict__ out) {
    out[0] = *(const float*)(ws + WS_ACCUM) * (1.0f / (float)TOTAL);
}

extern "C" void kernel_launch(void* const* d_in, const int* in_sizes, int n_in,
                              void* d_out, int out_size, void* d_ws, size_t ws_size,
                              hipStream_t stream) {
    const float* inp  = (const float*)d_in[0];   // input_data  (1,3,H,H)
    const float* tgt  = (const float*)d_in[1];   // target_data (1,3,H,H)
    const float* msrc = (const float*)d_in[2];   // mask_src    (1,1,H,H)
    const float* mtar = (const float*)d_in[3];   // mask_tar    (1,1,H,H)
    const int*   i0   = (const int*)d_in[4];
    const int*   i1   = (const int*)d_in[5];
    const int*   i2   = (const int*)d_in[6];
    const int*   i3   = (const int*)d_in[7];
    const float* ref  = (const float*)d_in[8];   // ref         (1,3,H,H)
    float* out = (float*)d_out;
    unsigned* ws = (unsigned*)d_ws;

    hm_zero_ws<<<512, 256, 0, stream>>>(ws);
    hm_hist  <<<2048, 256, 0, stream>>>(ref, tgt, msrc, mtar, i0, i1, i2, i3, ws);
    hm_table <<<1, BINS, 0, stream>>>(ws);
    hm_flags <<<2048, 256, 0, stream>>>(i0, i1, ws);
    hm_loss  <<<1536, 256, 0, stream>>>(inp, ref, msrc, ws);
    hm_final <<<1, 1, 0, stream>>>(ws, out);
}